// RITS_25469156065485
// MI455X (gfx1250) — compile-verified
//
#include <hip/hip_runtime.h>
#include <hip/hip_bf16.h>

typedef __attribute__((ext_vector_type(2))) float v2f;
typedef __attribute__((ext_vector_type(8))) float v8f;

#define B_ 256
#define L_ 256
#define K_ 128
#define H_ 256
#define BT 16          // batch rows per workgroup (one WMMA M-tile)
#define NT 256         // threads per workgroup
#define NW 8           // waves per workgroup (wave32)

// LDS row strides, chosen so stride mod 64 == 4 (dwords) -> conflict-free
// strided A-fragment b64 loads across the 16 rows of a WMMA tile.
#define LDK 132        // rows of length 128
#define LDH 260        // rows of length 256
#define LDG 772        // rows of length 768

struct alignas(16) SharedMem {
    float h_prev[BT][LDH];   // carried hidden state (pre-decay)
    float h_dec [BT][LDH];   // h * gamma_h (decayed)
    float cat1  [BT][LDH];   // [gamma_x | mt]
    float cat2  [BT][LDH];   // [c_c | mt]
    float gi    [BT][LDG];   // inputs @ W_ih^T + b_ih
    float gh    [BT][LDG];   // h_dec @ W_hh^T + b_hh
    float dprev [BT][LDK];   // unmasked delta carry
    float dtm   [BT][LDK];   // masked delta (GEMM A operand)
    float xt    [BT][LDK];
    float mt    [BT][LDK];
    float xh    [BT][LDK];
    float zh    [BT][LDK];
    float al    [BT][LDK];
    float xc    [BT][LDK];
    float red   [NT];
    float ebuf  [BT];
    float gapv  [BT];
    int   recn  [BT];
};

// One 16x16 f32 output tile: D = A(16xKd, LDS) * W^T(tile) accumulated with
// V_WMMA_F32_16X16X4_F32.  W is row-major [N][ldw]; B-frag[k][n] = W[n*ldw+k].
__device__ __forceinline__ v8f gemm16(const float* A, int lda,
                                      const float* __restrict__ W, int ldw,
                                      int nb, int Kd) {
    const int lane = threadIdx.x & 31;
    const int arow = lane & 15;              // M index for A frag
    const int koff = (lane >> 4) << 1;       // lanes 16-31 hold K=2,3 of each quad
    const float* ap = A + arow * lda + koff;
    const float* wp = W + (size_t)(nb + (lane & 15)) * ldw + koff;
    v8f c = {};
#pragma unroll 4
    for (int k = 0; k < Kd; k += 4) {
        v2f a = *(const v2f*)ap;
        v2f b = *(const v2f*)wp;
        c = __builtin_amdgcn_wmma_f32_16x16x4_f32(false, a, false, b,
                                                  (short)0, c, false, false);
        ap += 4; wp += 4;
    }
    return c;
}

// Two N-tiles sharing one A-fragment stream: halves DS traffic per WMMA.
__device__ __forceinline__ void gemm16x2(const float* A, int lda,
                                         const float* __restrict__ W, int ldw,
                                         int nb0, int nb1, int Kd,
                                         v8f& c0, v8f& c1) {
    const int lane = threadIdx.x & 31;
    const int arow = lane & 15;
    const int koff = (lane >> 4) << 1;
    const float* ap  = A + arow * lda + koff;
    const float* wp0 = W + (size_t)(nb0 + (lane & 15)) * ldw + koff;
    const float* wp1 = W + (size_t)(nb1 + (lane & 15)) * ldw + koff;
    c0 = (v8f){};
    c1 = (v8f){};
#pragma unroll 4
    for (int k = 0; k < Kd; k += 4) {
        v2f a  = *(const v2f*)ap;
        v2f b0 = *(const v2f*)wp0;
        v2f b1 = *(const v2f*)wp1;
        c0 = __builtin_amdgcn_wmma_f32_16x16x4_f32(false, a, false, b0,
                                                   (short)0, c0, false, false);
        c1 = __builtin_amdgcn_wmma_f32_16x16x4_f32(false, a, false, b1,
                                                   (short)0, c1, false, false);
        ap += 4; wp0 += 4; wp1 += 4;
    }
}

__device__ __forceinline__ float sigmoidf_(float v) {
    return 1.f / (1.f + expf(-v));
}

// Pass 1: inv_msum[t] = 1 / (sum_{b,k} mask[b,t,k] + 1e-5)
__global__ void rits_prep(const float* __restrict__ mask,
                          float* __restrict__ inv_msum) {
    __shared__ float red[NT];
    const int t = blockIdx.x;
    float s = 0.f;
    for (int i = threadIdx.x; i < B_ * K_; i += NT) {
        int b = i >> 7, k = i & 127;
        s += mask[(size_t)b * L_ * K_ + (size_t)t * K_ + k];
    }
    red[threadIdx.x] = s;
    __syncthreads();
    for (int off = NT / 2; off > 0; off >>= 1) {
        if (threadIdx.x < off) red[threadIdx.x] += red[threadIdx.x + off];
        __syncthreads();
    }
    if (threadIdx.x == 0) inv_msum[t] = 1.f / (red[0] + 1e-5f);
}

// Pass 2: full recurrence.  One workgroup = 16 batch rows, loops over all L.
__global__ void __launch_bounds__(NT) rits_main(
    const float* __restrict__ x, const float* __restrict__ mask,
    const float* __restrict__ ts,
    const float* __restrict__ W_gh, const float* __restrict__ b_gh,
    const float* __restrict__ W_gx, const float* __restrict__ b_gx,
    const float* __restrict__ W_hist, const float* __restrict__ b_hist,
    const float* __restrict__ W_feat, const float* __restrict__ b_feat,
    const float* __restrict__ W_comb, const float* __restrict__ b_comb,
    const float* __restrict__ W_ih, const float* __restrict__ W_hh,
    const float* __restrict__ b_ih, const float* __restrict__ b_hh,
    const float* __restrict__ W_out, const float* __restrict__ b_out,
    const int* __restrict__ record_num, const float* __restrict__ inv_msum,
    float* __restrict__ imp_out, float* __restrict__ pred_out,
    float* __restrict__ loss_partial)
{
    __shared__ SharedMem s;
    const int tid   = threadIdx.x;
    const int wave  = tid >> 5;
    const int lane  = tid & 31;
    const int col   = lane & 15;
    const int rbase = (lane >> 4) << 3;      // C/D frag: lanes 16-31 hold M=8..15
    const int b0    = blockIdx.x * BT;
    float loss_acc = 0.f;

    for (int i = tid; i < BT * H_; i += NT) s.h_prev[i >> 8][i & 255] = 0.f;
    if (tid < BT) s.recn[tid] = record_num[b0 + tid];
    __syncthreads();

    for (int t = 0; t < L_; ++t) {
        // ---- P1: per-row scalars, load x/m, delta recurrence, gamma_x ----
        if (tid < BT) {
            float tv = ts[(size_t)(b0 + tid) * L_ + t];
            s.ebuf[tid] = (tv != 0.f) ? 1.f : 0.f;
            s.gapv[tid] = (t > 0) ? fabsf(tv - ts[(size_t)(b0 + tid) * L_ + t - 1]) : 0.f;
        }
        __syncthreads();
        for (int i = tid; i < BT * K_; i += NT) {
            int m = i >> 7, k = i & 127;
            size_t g = (size_t)(b0 + m) * L_ * K_ + (size_t)t * K_ + k;
            float xv = x[g], mv = mask[g];
            if (t + 1 < L_) {               // warm next step's streams (HBM)
                __builtin_prefetch(&x[g + K_], 0, 0);
                __builtin_prefetch(&mask[g + K_], 0, 0);
            }
            s.xt[m][k] = xv;
            s.mt[m][k] = mv;
            float d = (t == 0) ? 1.f : (s.gapv[m] + (1.f - mv) * s.dprev[m][k]);
            s.dprev[m][k] = d;
            float dm = (t < s.recn[m]) ? d : 0.f;
            s.dtm[m][k] = dm;
            // gamma_x: diagonal W_gx only
            float gx = expf(-fmaxf(dm * W_gx[k * K_ + k] + b_gx[k], 0.f));
            s.cat1[m][k]      = gx;
            s.cat1[m][K_ + k] = mv;
            s.cat2[m][K_ + k] = mv;
        }
        __syncthreads();

        // ---- P2: gamma_h = exp(-relu(dtm @ W_gh^T + b)); h_dec = h*gamma ----
        // 16 tiles -> 8 paired tasks, one per wave.
        {
            int nb0 = wave * 32, nb1 = nb0 + 16;
            v8f a0, a1;
            gemm16x2(&s.dtm[0][0], LDK, W_gh, K_, nb0, nb1, K_, a0, a1);
            int n0 = nb0 + col, n1 = nb1 + col;
            float bi0 = b_gh[n0], bi1 = b_gh[n1];
#pragma unroll
            for (int v = 0; v < 8; ++v) {
                int r = rbase + v;
                float g0 = expf(-fmaxf(a0[v] + bi0, 0.f));
                float g1 = expf(-fmaxf(a1[v] + bi1, 0.f));
                s.h_dec[r][n0] = s.h_prev[r][n0] * g0;
                s.h_dec[r][n1] = s.h_prev[r][n1] * g1;
            }
        }
        __syncthreads();

        // ---- P3: x_h = h_dec @ W_hist^T + b_hist (8 tiles, 1 per wave) ----
        {
            int nb = wave * 16;
            v8f acc = gemm16(&s.h_dec[0][0], LDH, W_hist, H_, nb, H_);
            int n = nb + col;
            float bias = b_hist[n];
#pragma unroll
            for (int v = 0; v < 8; ++v) s.xh[rbase + v][n] = acc[v] + bias;
        }
        __syncthreads();

        // ---- P4: loss1, x_c ----
        const float im = inv_msum[t];
        {
            float l1 = 0.f;
            for (int i = tid; i < BT * K_; i += NT) {
                int m = i >> 7, k = i & 127;
                float xv = s.xt[m][k], mv = s.mt[m][k], xhv = s.xh[m][k];
                l1 += fabsf(xv - xhv) * mv;
                s.xc[m][k] = mv * xv + (1.f - mv) * xhv;
            }
            loss_acc += l1 * im;
        }
        __syncthreads();

        // ---- P5: z_h (zero-diag W_feat) and alpha; 8 paired tasks ----
        if (wave < 4) {
            int nb0 = wave * 32, nb1 = nb0 + 16;
            v8f a0, a1;
            gemm16x2(&s.xc[0][0], LDK, W_feat, K_, nb0, nb1, K_, a0, a1);
            int n0 = nb0 + col, n1 = nb1 + col;
            float bi0 = b_feat[n0], bi1 = b_feat[n1];
            float wd0 = W_feat[n0 * K_ + n0], wd1 = W_feat[n1 * K_ + n1];
#pragma unroll
            for (int v = 0; v < 8; ++v) {
                int r = rbase + v;
                s.zh[r][n0] = a0[v] + bi0 - wd0 * s.xc[r][n0];
                s.zh[r][n1] = a1[v] + bi1 - wd1 * s.xc[r][n1];
            }
        } else {
            int nb0 = (wave - 4) * 32, nb1 = nb0 + 16;
            v8f a0, a1;
            gemm16x2(&s.cat1[0][0], LDH, W_comb, 2 * K_, nb0, nb1, 2 * K_, a0, a1);
            int n0 = nb0 + col, n1 = nb1 + col;
            float bi0 = b_comb[n0], bi1 = b_comb[n1];
#pragma unroll
            for (int v = 0; v < 8; ++v) {
                s.al[rbase + v][n0] = a0[v] + bi0;
                s.al[rbase + v][n1] = a1[v] + bi1;
            }
        }
        __syncthreads();

        // ---- P7: loss2/3, c_h, c_c, imputation output, cat2 left half ----
        {
            float l23 = 0.f;
            for (int i = tid; i < BT * K_; i += NT) {
                int m = i >> 7, k = i & 127;
                float xv = s.xt[m][k], mv = s.mt[m][k];
                float e = s.ebuf[m];
                float zv = s.zh[m][k], av = s.al[m][k], xhv = s.xh[m][k];
                l23 += fabsf(xv - zv) * mv * e;
                float ch = av * zv + (1.f - av) * xhv;
                l23 += fabsf(xv - ch) * mv * e;
                float cc = mv * xv + (1.f - mv) * ch;
                s.cat2[m][k] = cc;
                // streaming (non-temporal) store: write-once output
                __builtin_nontemporal_store(
                    cc * e,
                    &imp_out[(size_t)(b0 + m) * L_ * K_ + (size_t)t * K_ + k]);
            }
            loss_acc += l23 * im;
        }
        __syncthreads();

        // ---- P8: gi = cat2 @ W_ih^T + b_ih, gh = h_dec @ W_hh^T + b_hh ----
        // 96 tiles -> 48 paired tasks, 6 per wave.
        for (int task = wave; task < 48; task += NW) {
            if (task < 24) {
                int nb0 = task * 32, nb1 = nb0 + 16;
                v8f a0, a1;
                gemm16x2(&s.cat2[0][0], LDH, W_ih, 2 * K_, nb0, nb1, 2 * K_, a0, a1);
                int n0 = nb0 + col, n1 = nb1 + col;
                float bi0 = b_ih[n0], bi1 = b_ih[n1];
#pragma unroll
                for (int v = 0; v < 8; ++v) {
                    s.gi[rbase + v][n0] = a0[v] + bi0;
                    s.gi[rbase + v][n1] = a1[v] + bi1;
                }
            } else {
                int nb0 = (task - 24) * 32, nb1 = nb0 + 16;
                v8f a0, a1;
                gemm16x2(&s.h_dec[0][0], LDH, W_hh, H_, nb0, nb1, H_, a0, a1);
                int n0 = nb0 + col, n1 = nb1 + col;
                float bi0 = b_hh[n0], bi1 = b_hh[n1];
#pragma unroll
                for (int v = 0; v < 8; ++v) {
                    s.gh[rbase + v][n0] = a0[v] + bi0;
                    s.gh[rbase + v][n1] = a1[v] + bi1;
                }
            }
        }
        __syncthreads();

        // ---- P9: GRU cell + validity gating ----
        for (int i = tid; i < BT * H_; i += NT) {
            int m = i >> 8, j = i & 255;
            float ir = s.gi[m][j], iz = s.gi[m][j + H_], ig = s.gi[m][j + 2 * H_];
            float hr = s.gh[m][j], hz = s.gh[m][j + H_], hg = s.gh[m][j + 2 * H_];
            float r = sigmoidf_(ir + hr);
            float z = sigmoidf_(iz + hz);
            float n = tanhf(ig + r * hg);
            float hn = (1.f - z) * n + z * s.h_dec[m][j];
            float e = s.ebuf[m];
            s.h_prev[m][j] = e * hn + (1.f - e) * s.h_prev[m][j];
        }
        __syncthreads();
    }

    // predictions = sigmoid(h_fin @ W_out^T + b_out)
    if (tid < BT) {
        float acc = 0.f;
        for (int j = 0; j < H_; ++j) acc += s.h_prev[tid][j] * W_out[j];
        pred_out[b0 + tid] = sigmoidf_(acc + b_out[0]);
    }

    // deterministic per-WG loss partial
    s.red[tid] = loss_acc;
    __syncthreads();
    for (int off = NT / 2; off > 0; off >>= 1) {
        if (tid < off) s.red[tid] += s.red[tid + off];
        __syncthreads();
    }
    if (tid == 0) loss_partial[blockIdx.x] = s.red[0];
}

// Pass 3: deterministic sum of per-WG loss partials.
__global__ void rits_fin(const float* __restrict__ partials,
                         float* __restrict__ loss_out) {
    if (threadIdx.x == 0) {
        float s = 0.f;
        for (int i = 0; i < B_ / BT; ++i) s += partials[i];
        *loss_out = s;
    }
}

extern "C" void kernel_launch(void* const* d_in, const int* in_sizes, int n_in,
                              void* d_out, int out_size, void* d_ws, size_t ws_size,
                              hipStream_t stream) {
    const float* x         = (const float*)d_in[0];
    const float* mask      = (const float*)d_in[1];
    const float* ts        = (const float*)d_in[2];
    const float* W_gh      = (const float*)d_in[3];
    const float* b_gh      = (const float*)d_in[4];
    const float* W_gx      = (const float*)d_in[5];
    const float* b_gx      = (const float*)d_in[6];
    const float* W_hist    = (const float*)d_in[7];
    const float* b_hist    = (const float*)d_in[8];
    const float* W_feat    = (const float*)d_in[9];
    const float* b_feat    = (const float*)d_in[10];
    const float* W_comb    = (const float*)d_in[11];
    const float* b_comb    = (const float*)d_in[12];
    const float* W_ih      = (const float*)d_in[13];
    const float* W_hh      = (const float*)d_in[14];
    const float* b_ih      = (const float*)d_in[15];
    const float* b_hh      = (const float*)d_in[16];
    const float* W_out     = (const float*)d_in[17];
    const float* b_out     = (const float*)d_in[18];
    const int*   record_nm = (const int*)d_in[19];

    float* out  = (float*)d_out;
    float* imp  = out;                              // B*L*K imputations
    float* loss = out + (size_t)B_ * L_ * K_;       // 1 scalar
    float* pred = loss + 1;                         // B predictions

    float* invm     = (float*)d_ws;                 // L floats
    float* partials = invm + L_;                    // B_/BT floats

    rits_prep<<<L_, NT, 0, stream>>>(mask, invm);
    rits_main<<<B_ / BT, NT, 0, stream>>>(
        x, mask, ts, W_gh, b_gh, W_gx, b_gx, W_hist, b_hist, W_feat, b_feat,
        W_comb, b_comb, W_ih, W_hh, b_ih, b_hh, W_out, b_out,
        record_nm, invm, imp, pred, partials);
    rits_fin<<<1, 32, 0, stream>>>(partials, loss);
}